// SurfaceDilationLogic_13426067767996
// MI455X (gfx1250) — compile-verified
//
#include <hip/hip_runtime.h>
#include <stdint.h>

// ---------------------------------------------------------------------------
// SurfaceDilationLogic for MI455X (gfx1250, wave32)
// boundary = p - minpool3^3(p); dilated = maxpool(9,9,5); soft-dice-like loss.
// Memory-bound stencil: separable pooling + async global->LDS staging (CDNA5
// async-tensor path) + WMMA-based wave32 reduction for the final sums.
// ---------------------------------------------------------------------------

typedef float v2f __attribute__((ext_vector_type(2)));
typedef float v4f __attribute__((ext_vector_type(4)));
typedef float v8f __attribute__((ext_vector_type(8)));

constexpr int DIM  = 128;
constexpr int SLAB = DIM * DIM * DIM;   // 2,097,152 elements per (b,c)
constexpr int NBC  = 6;                 // 2*3 slabs
#define NEG_HUGE (-3.402823466e+38f)

__device__ __forceinline__ int clampi(int v, int lo, int hi) {
    return v < lo ? lo : (v > hi ? hi : v);
}

__device__ __forceinline__ v4f max4(v4f a, v4f b) {
    v4f r;
    r.x = fmaxf(a.x, b.x); r.y = fmaxf(a.y, b.y);
    r.z = fmaxf(a.z, b.z); r.w = fmaxf(a.w, b.w);
    return r;
}

// CDNA5 async global->LDS copy (per-lane, 4B). LDS offset is relative to the
// workgroup LDS base; our single __shared__ block starts at offset 0.
__device__ __forceinline__ void async_ld_lds_b32(uint32_t lds_byte_off, const float* gaddr) {
    asm volatile("global_load_async_to_lds_b32 %0, %1, off"
                 :: "v"(lds_byte_off), "v"(gaddr) : "memory");
}
__device__ __forceinline__ void wait_async_all() {
    asm volatile("s_wait_asynccnt 0" ::: "memory");
}

// ---------------------------------------------------------------------------
// K1: boundary = p - minpool3x3x3(p), SAME (clamped) windows.
// Tile 4(y) x 8(x) x 32(z), halo 1 -> 6x10x34 staged into LDS via async loads,
// then separable z/x/y mins inside LDS.
// Grid: NBC * 32 * 16 * 4 = 12288 blocks, 256 threads.
// ---------------------------------------------------------------------------
__global__ __launch_bounds__(256)
void boundary_kernel(const float* __restrict__ in, float* __restrict__ out) {
    constexpr int TY = 4, TX = 8, TZ = 32;
    constexpr int IY = TY + 2, IX = TX + 2, IZ = TZ + 2;   // 6,10,34
    constexpr int NIN = IY * IX * IZ;                      // 2040
    constexpr int NZM = IY * IX * TZ;                      // 1920
    constexpr int NXZ = IY * TX * TZ;                      // 1536
    __shared__ float ls[NIN + NZM + NXZ];                  // ~22 KB
    float* lin = ls;
    float* lzm = ls + NIN;
    float* lxz = ls + NIN + NZM;

    const int tid = threadIdx.x;
    int b = blockIdx.x;
    const int zt = b & 3;  b >>= 2;
    const int xt = b & 15; b >>= 4;
    const int yt = b & 31; b >>= 5;
    const int bc = b;
    const int y0 = yt * TY, x0 = xt * TX, z0 = zt * TZ;
    const float* src = in + (size_t)bc * SLAB;

    // Stage halo tile into LDS with async copies (clamped indices = SAME pad).
    for (int i = tid; i < NIN; i += 256) {
        int iz = i % IZ; int r = i / IZ;
        int ix = r % IX; int iy = r / IX;
        const int gy = clampi(y0 + iy - 1, 0, DIM - 1);
        const int gx = clampi(x0 + ix - 1, 0, DIM - 1);
        const int gz = clampi(z0 + iz - 1, 0, DIM - 1);
        const float* ga = src + ((size_t)(gy * DIM + gx) * DIM + gz);
        async_ld_lds_b32((uint32_t)(i * 4), ga);
    }
    wait_async_all();
    __syncthreads();

    // min over z (width 3)
    for (int i = tid; i < NZM; i += 256) {
        int zl = i % TZ; int r = i / TZ;
        int ix = r % IX; int iy = r / IX;
        const float* p = lin + (iy * IX + ix) * IZ + zl;
        lzm[i] = fminf(fminf(p[0], p[1]), p[2]);
    }
    __syncthreads();

    // min over x (width 3)
    for (int i = tid; i < NXZ; i += 256) {
        int zl = i % TZ; int r = i / TZ;
        int xl = r % TX; int iy = r / TX;
        const float* p = lzm + (iy * IX + xl) * TZ + zl;
        lxz[i] = fminf(fminf(p[0], p[TZ]), p[2 * TZ]);
    }
    __syncthreads();

    // min over y (width 3), subtract from center, store boundary
    for (int i = tid; i < TY * TX * TZ; i += 256) {
        int zl = i % TZ; int r = i / TZ;
        int xl = r % TX; int yl = r / TX;
        const float* p = lxz + (yl * TX + xl) * TZ + zl;
        const float m = fminf(fminf(p[0], p[TX * TZ]), p[2 * TX * TZ]);
        const float c = lin[((yl + 1) * IX + (xl + 1)) * IZ + (zl + 1)];
        out[(size_t)bc * SLAB + ((size_t)(y0 + yl) * DIM + (x0 + xl)) * DIM + (z0 + zl)] = c - m;
    }
}

// ---------------------------------------------------------------------------
// K2: max over z (width 5) then x (width 9) per y-plane.
// Tile 16(x) x 64(z), halo 4(x)/2(z) -> 24x68 staged via async loads.
// Grid: NBC * 128 * 8 * 2 = 12288 blocks, 256 threads.
// ---------------------------------------------------------------------------
__global__ __launch_bounds__(256)
void dilate_zx_kernel(const float* __restrict__ in, float* __restrict__ out) {
    constexpr int TX = 16, TZ = 64;
    constexpr int IX = TX + 8, IZ = TZ + 4;    // 24, 68
    constexpr int NIN = IX * IZ;               // 1632
    constexpr int NZM = IX * TZ;               // 1536
    __shared__ float ls[NIN + NZM];            // ~12.7 KB
    float* lin = ls;
    float* lzm = ls + NIN;

    const int tid = threadIdx.x;
    int b = blockIdx.x;
    const int zt = b & 1;   b >>= 1;
    const int xt = b & 7;   b >>= 3;
    const int y  = b & 127; b >>= 7;
    const int bc = b;
    const int x0 = xt * TX, z0 = zt * TZ;
    const float* src = in + (size_t)bc * SLAB + (size_t)y * DIM * DIM;

    for (int i = tid; i < NIN; i += 256) {
        int iz = i % IZ; int ix = i / IZ;
        const int gx = clampi(x0 + ix - 4, 0, DIM - 1);
        const int gz = clampi(z0 + iz - 2, 0, DIM - 1);
        async_ld_lds_b32((uint32_t)(i * 4), src + (size_t)gx * DIM + gz);
    }
    wait_async_all();
    __syncthreads();

    // max over z (width 5)
    for (int i = tid; i < NZM; i += 256) {
        int zl = i % TZ; int ix = i / TZ;
        const float* p = lin + ix * IZ + zl;
        float m = p[0];
        m = fmaxf(m, p[1]); m = fmaxf(m, p[2]);
        m = fmaxf(m, p[3]); m = fmaxf(m, p[4]);
        lzm[i] = m;
    }
    __syncthreads();

    // max over x (width 9)
    for (int i = tid; i < TX * TZ; i += 256) {
        int zl = i % TZ; int xl = i / TZ;
        const float* p = lzm + xl * TZ + zl;
        float m = p[0];
#pragma unroll
        for (int d = 1; d < 9; ++d) m = fmaxf(m, p[d * TZ]);
        out[(size_t)bc * SLAB + (size_t)y * DIM * DIM + (size_t)(x0 + xl) * DIM + (z0 + zl)] = m;
    }
}

// ---------------------------------------------------------------------------
// Wave32 sum-reduction through V_WMMA_F32_16X16X4_F32:
// A(16x4) holds lane values (K0 = lanes 0..15, K2 = lanes 16..31; K1/K3 = 0),
// B = ones -> D[m][n] = v_m + v_{m+16}. Summing the 8 D VGPRs gives each lane
// half the total; one xor-16 shuffle completes the 32-lane sum.
// ---------------------------------------------------------------------------
__device__ __forceinline__ float wave_reduce_wmma(float v) {
    v2f a; a.x = v;    a.y = 0.0f;
    v2f o; o.x = 1.0f; o.y = 1.0f;
    v8f c = {};
    v8f d = __builtin_amdgcn_wmma_f32_16x16x4_f32(false, a, false, o,
                                                  (short)0, c, false, false);
    float r = d[0] + d[1] + d[2] + d[3] + d[4] + d[5] + d[6] + d[7];
    r += __shfl_xor(r, 16, 32);
    return r;
}

// ---------------------------------------------------------------------------
// K3: max over y (width 9) for both tensors + fused reductions.
// One thread per (bc, x, 4-wide z group): float4 (b128) streaming loads with
// NT hint (each dilated value is consumed exactly once), 9-deep register
// sliding window over y, vector accumulators folded once at the end.
// Grid: 6*128*32/256 = 96 blocks; EXEC all-ones (required for WMMA).
// ---------------------------------------------------------------------------
__global__ __launch_bounds__(256)
void dilate_y_reduce_kernel(const float* __restrict__ dp,
                            const float* __restrict__ dt,
                            float* __restrict__ acc) {
    int t = blockIdx.x * 256 + threadIdx.x;
    const int zg = t & 31;  t >>= 5;   // 4-element z group
    const int x  = t & 127; t >>= 7;
    const int bc = t;
    const size_t col = (size_t)bc * SLAB + (size_t)x * DIM + (size_t)zg * 4;
    const v4f* P = (const v4f*)(dp + col);
    const v4f* T = (const v4f*)(dt + col);
    constexpr int YS4 = DIM * DIM / 4;  // float4 stride per y step

    const v4f ninf = { NEG_HUGE, NEG_HUGE, NEG_HUGE, NEG_HUGE };
    v4f wp[9], wt[9];
#pragma unroll
    for (int i = 0; i < 9; ++i) { wp[i] = ninf; wt[i] = ninf; }
#pragma unroll
    for (int i = 0; i < 5; ++i) {
        wp[4 + i] = __builtin_nontemporal_load(P + (size_t)i * YS4);
        wt[4 + i] = __builtin_nontemporal_load(T + (size_t)i * YS4);
    }

    v4f sp = {0.f, 0.f, 0.f, 0.f};
    v4f st = {0.f, 0.f, 0.f, 0.f};
    v4f spt = {0.f, 0.f, 0.f, 0.f};
    for (int y = 0; y < DIM; ++y) {
        v4f mp = wp[0], mt = wt[0];
#pragma unroll
        for (int i = 1; i < 9; ++i) { mp = max4(mp, wp[i]); mt = max4(mt, wt[i]); }
        sp += mp; st += mt; spt += mp * mt;
#pragma unroll
        for (int i = 0; i < 8; ++i) { wp[i] = wp[i + 1]; wt[i] = wt[i + 1]; }
        const int yn = y + 5;
        if (yn < DIM) {
            wp[8] = __builtin_nontemporal_load(P + (size_t)yn * YS4);
            wt[8] = __builtin_nontemporal_load(T + (size_t)yn * YS4);
        } else {
            wp[8] = ninf; wt[8] = ninf;
        }
    }

    const float fsp  = (sp.x + sp.y) + (sp.z + sp.w);
    const float fst  = (st.x + st.y) + (st.z + st.w);
    const float fspt = (spt.x + spt.y) + (spt.z + spt.w);

    const float rp  = wave_reduce_wmma(fsp);
    const float rt  = wave_reduce_wmma(fst);
    const float rpt = wave_reduce_wmma(fspt);
    if ((threadIdx.x & 31) == 0) {
        atomicAdd(&acc[bc * 4 + 0], rp);
        atomicAdd(&acc[bc * 4 + 1], rt);
        atomicAdd(&acc[bc * 4 + 2], rpt);
    }
}

// ---------------------------------------------------------------------------
// K0 / K4: accumulator init + scalar epilogue.
// ---------------------------------------------------------------------------
__global__ void zero_acc_kernel(float* acc) {
    if (threadIdx.x < 32) acc[threadIdx.x] = 0.0f;
}

__global__ void finalize_kernel(const float* __restrict__ acc, float* __restrict__ out) {
    if (threadIdx.x == 0 && blockIdx.x == 0) {
        float m = 0.0f;
        for (int bc = 0; bc < NBC; ++bc) {
            const float ov = acc[bc * 4 + 2];
            const float un = acc[bc * 4 + 0] + acc[bc * 4 + 1];
            m += 2.0f * ov / (un + 1e-5f);
        }
        out[0] = 1.0f - m / (float)NBC;
    }
}

// ---------------------------------------------------------------------------
// Host entry. Inputs: d_in[0]=y_true, d_in[1]=y_pred, d_in[2]=exclude_background
// (==1, so no channel slicing per the reference). Output: 1 float.
// Workspace: 64 accumulator floats + 3 slab buffers (~151 MB).
// ---------------------------------------------------------------------------
extern "C" void kernel_launch(void* const* d_in, const int* in_sizes, int n_in,
                              void* d_out, int out_size, void* d_ws, size_t ws_size,
                              hipStream_t stream) {
    (void)in_sizes; (void)n_in; (void)out_size; (void)ws_size;
    const float* y_true = (const float*)d_in[0];
    const float* y_pred = (const float*)d_in[1];
    float* out = (float*)d_out;

    float* acc   = (float*)d_ws;                        // 64 floats
    float* bufA  = acc + 64;                            // NBC*SLAB (shared scratch)
    float* dil_p = bufA + (size_t)NBC * SLAB;           // NBC*SLAB
    float* dil_t = dil_p + (size_t)NBC * SLAB;          // NBC*SLAB

    const int stencil_blocks = NBC * 32 * 16 * 4;       // 12288 (both K1 and K2)

    zero_acc_kernel<<<1, 32, 0, stream>>>(acc);

    boundary_kernel <<<stencil_blocks, 256, 0, stream>>>(y_pred, bufA);
    dilate_zx_kernel<<<stencil_blocks, 256, 0, stream>>>(bufA, dil_p);

    boundary_kernel <<<stencil_blocks, 256, 0, stream>>>(y_true, bufA);
    dilate_zx_kernel<<<stencil_blocks, 256, 0, stream>>>(bufA, dil_t);

    dilate_y_reduce_kernel<<<(NBC * DIM * DIM / 4) / 256, 256, 0, stream>>>(dil_p, dil_t, acc);
    finalize_kernel<<<1, 1, 0, stream>>>(acc, out);
}